// RectConv2d_32049045963395
// MI455X (gfx1250) — compile-verified
//
#include <hip/hip_runtime.h>
#include <hip/hip_bf16.h>
#include <math.h>

typedef __attribute__((ext_vector_type(2))) float v2f;
typedef __attribute__((ext_vector_type(8))) float v8f;

#define BSZ   4
#define CIN   64
#define OUTC  64
#define HH    128
#define WW    128
#define HW    (HH*WW)
#define KC    576          // CIN * 9
#define ASTRIDE 292        // 288 + 4 pad -> conflict-free b64 LDS reads

__device__ __forceinline__ float sigmoidf_(float z) { return 1.0f / (1.0f + __expf(-z)); }

// -------------------------------------------------------------------------
// Kernel 1: 12-channel 3x3 conv -> l, wd, cos(theta), sin(theta), mod[9]
// One thread per pixel. Weight indices are wave-uniform -> scalar loads.
// ws planes (each BSZ*HW floats): 0=l 1=wd 2=ct 3=st 4..12=mod[n]
// -------------------------------------------------------------------------
__global__ __launch_bounds__(128) void rect_offsets_kernel(
    const float* __restrict__ x,
    const float* __restrict__ w_p, const float* __restrict__ b_p,
    const float* __restrict__ w_m, const float* __restrict__ b_m,
    float* __restrict__ ws)
{
    const int pix = blockIdx.x * 128 + threadIdx.x;
    const int b = pix >> 14;
    const int h = (pix >> 7) & 127;
    const int w = pix & 127;
    const float* xb = x + b * CIN * HW;

    float sp0 = 0.f, sp1 = 0.f, sp2 = 0.f;
    float sm[9];
#pragma unroll
    for (int n = 0; n < 9; ++n) sm[n] = 0.f;

    for (int c = 0; c < CIN; ++c) {
        const float* xc = xb + c * HW;
        float v[9];
#pragma unroll
        for (int t = 0; t < 9; ++t) {
            const int i = t / 3, j = t % 3;
            const int rr = h + i - 1, cc = w + j - 1;
            v[t] = (rr >= 0 && rr < HH && cc >= 0 && cc < WW) ? xc[rr * WW + cc] : 0.f;
        }
        const int kb = c * 9;
#pragma unroll
        for (int t = 0; t < 9; ++t) {
            sp0 = fmaf(w_p[0 * KC + kb + t], v[t], sp0);
            sp1 = fmaf(w_p[1 * KC + kb + t], v[t], sp1);
            sp2 = fmaf(w_p[2 * KC + kb + t], v[t], sp2);
#pragma unroll
            for (int n = 0; n < 9; ++n)
                sm[n] = fmaf(w_m[n * KC + kb + t], v[t], sm[n]);
        }
    }

    const float l  = sigmoidf_(__sinf(sp0 + b_p[0])) * 48.5f + 1.5f;
    const float wd = sigmoidf_(__sinf(sp1 + b_p[1])) * 48.5f + 1.5f;
    const float th = sigmoidf_(__sinf(sp2 + b_p[2])) * 3.1415926f;

    const int P = BSZ * HW;
    ws[0 * P + pix] = l;
    ws[1 * P + pix] = wd;
    ws[2 * P + pix] = __cosf(th);
    ws[3 * P + pix] = __sinf(th);
#pragma unroll
    for (int n = 0; n < 9; ++n)
        ws[(4 + n) * P + pix] = sigmoidf_(sm[n] + b_m[n]);
}

// -------------------------------------------------------------------------
// Kernel 2: fused deformable-gather + double GEMM (w_c over sampled A,
// w_b over im2col A) via V_WMMA_F32_16X16X4_F32, plus b_b bias.
// Tile: 16 pixels (one row segment) x 64 outc. 4 waves, wave wv -> outc
// [16wv,16wv+16). K = 1152 split into 4 chunks of 288 staged in LDS.
// -------------------------------------------------------------------------
__global__ __launch_bounds__(128) void rect_main_kernel(
    const float* __restrict__ x,
    const float* __restrict__ w_b, const float* __restrict__ b_b,
    const float* __restrict__ w_c,
    const float* __restrict__ ws,
    float* __restrict__ out)
{
    __shared__ float Alds[16 * ASTRIDE];
    __shared__ int   s_qx0[144], s_qy0[144], s_qx1[144], s_qy1[144];
    __shared__ float s_glt[144], s_grb[144], s_glb[144], s_grt[144], s_mod[144];

    const int tid  = threadIdx.x;
    const int lane = tid & 31;
    const int wave = tid >> 5;
    const int w0 = blockIdx.x * 16;
    const int h  = blockIdx.y;
    const int b  = blockIdx.z;
    const int P  = BSZ * HW;
    const int pixbase = (b * HH + h) * WW + w0;
    const float* xb = x + b * CIN * HW;

    // ---- stage 1: per-(pixel m, point n) sampling geometry -------------
    for (int id = tid; id < 144; id += 128) {
        const int m = id / 9, n = id % 9;
        const int pix = pixbase + m;
        const float l   = ws[0 * P + pix];
        const float wdv = ws[1 * P + pix];
        const float ct  = ws[2 * P + pix];
        const float st  = ws[3 * P + pix];
        const float pnx = (float)(n / 3 - 1);
        const float pny = (float)(n % 3 - 1);
        const float px = (float)(h + 1)      + (l   * (1.f / 3.f)) * pnx;
        const float py = (float)(w0 + m + 1) + (wdv * (1.f / 3.f)) * pny;
        const float pxx = px * ct - py * st;
        const float pyy = px * st + py * ct;
        const float fx = floorf(pxx), fy = floorf(pyy);
        const float q0x = fminf(fmaxf(fx,       0.f), 129.f);
        const float q0y = fminf(fmaxf(fy,       0.f), 129.f);
        const float q1x = fminf(fmaxf(fx + 1.f, 0.f), 129.f);
        const float q1y = fminf(fmaxf(fy + 1.f, 0.f), 129.f);
        const float pcx = fminf(fmaxf(pxx, 0.f), 129.f);
        const float pcy = fminf(fmaxf(pyy, 0.f), 129.f);
        s_qx0[id] = (int)q0x; s_qy0[id] = (int)q0y;
        s_qx1[id] = (int)q1x; s_qy1[id] = (int)q1y;
        s_glt[id] = (1.f + (q0x - pcx)) * (1.f + (q0y - pcy));
        s_grb[id] = (1.f - (q1x - pcx)) * (1.f - (q1y - pcy));
        s_glb[id] = (1.f + (q0x - pcx)) * (1.f - (q1y - pcy));
        s_grt[id] = (1.f - (q1x - pcx)) * (1.f + (q0y - pcy));
        s_mod[id] = ws[(4 + n) * P + pix];
    }

    v8f acc0 = {};
    v8f acc1 = {};
    const int mrow = lane & 15;
    const int kk   = (lane >> 4) * 2;   // K sub-offset for this half-wave

    for (int chunk = 0; chunk < 4; ++chunk) {
        const int cbase   = (chunk & 1) * 32;
        const bool deform = (chunk < 2);
        __syncthreads();   // previous chunk's WMMA reads done before rewrite

        // ---- build A chunk in LDS: 16 pixels x (32 ch * 9 pts) --------
        for (int id = tid; id < 16 * 288; id += 128) {
            const int cl = id / 144;     // local channel 0..31
            const int r  = id % 144;
            const int m  = r / 9;
            const int n  = r % 9;
            const float* xc = xb + (cbase + cl) * HW;
            float val;
            if (deform) {
                const int g = m * 9 + n;
                const int x0 = s_qx0[g], y0 = s_qy0[g];
                const int x1 = s_qx1[g], y1 = s_qy1[g];
                const float f00 = (x0 >= 1 && x0 <= HH && y0 >= 1 && y0 <= WW) ? xc[(x0 - 1) * WW + (y0 - 1)] : 0.f;
                const float f11 = (x1 >= 1 && x1 <= HH && y1 >= 1 && y1 <= WW) ? xc[(x1 - 1) * WW + (y1 - 1)] : 0.f;
                const float f01 = (x0 >= 1 && x0 <= HH && y1 >= 1 && y1 <= WW) ? xc[(x0 - 1) * WW + (y1 - 1)] : 0.f;
                const float f10 = (x1 >= 1 && x1 <= HH && y0 >= 1 && y0 <= WW) ? xc[(x1 - 1) * WW + (y0 - 1)] : 0.f;
                val = (s_glt[g] * f00 + s_grb[g] * f11 +
                       s_glb[g] * f01 + s_grt[g] * f10) * s_mod[g];
            } else {
                const int i = n / 3, j = n % 3;
                const int rr = h + i - 1;
                const int cc = w0 + m + j - 1;
                val = (rr >= 0 && rr < HH && cc >= 0 && cc < WW) ? xc[rr * WW + cc] : 0.f;
            }
            Alds[m * ASTRIDE + cl * 9 + n] = val;
        }
        __syncthreads();

        // ---- WMMA over this chunk's 288 K values ----------------------
        const float* wbase = deform ? w_c : w_b;
        const float* wp = wbase + (wave * 16 + (lane & 15)) * KC + cbase * 9 + kk;
        const float* ap = &Alds[mrow * ASTRIDE + kk];
#pragma unroll 4
        for (int k0 = 0; k0 < 288; k0 += 8) {
            v2f a0 = *(const v2f*)(ap + k0);
            v2f b0 = *(const v2f*)(wp + k0);
            acc0 = __builtin_amdgcn_wmma_f32_16x16x4_f32(
                false, a0, false, b0, (short)0, acc0, false, false);
            v2f a1 = *(const v2f*)(ap + k0 + 4);
            v2f b1 = *(const v2f*)(wp + k0 + 4);
            acc1 = __builtin_amdgcn_wmma_f32_16x16x4_f32(
                false, a1, false, b1, (short)0, acc1, false, false);
        }
    }

    const v8f acc = acc0 + acc1;
    const int o_lane = wave * 16 + (lane & 15);
    const float bias = b_b[o_lane];
    float* ob = out + ((b * OUTC + o_lane) * HH + h) * WW;
    const int wadd = w0 + ((lane >> 4) * 8);
#pragma unroll
    for (int r = 0; r < 8; ++r)
        ob[wadd + r] = acc[r] + bias;
}

// -------------------------------------------------------------------------
// Host launcher. Input order: x, w_b, b_b, w_p, b_p, w_m, b_m, w_c
// -------------------------------------------------------------------------
extern "C" void kernel_launch(void* const* d_in, const int* in_sizes, int n_in,
                              void* d_out, int out_size, void* d_ws, size_t ws_size,
                              hipStream_t stream) {
    (void)in_sizes; (void)n_in; (void)out_size; (void)ws_size;
    const float* x   = (const float*)d_in[0];
    const float* w_b = (const float*)d_in[1];
    const float* b_b = (const float*)d_in[2];
    const float* w_p = (const float*)d_in[3];
    const float* b_p = (const float*)d_in[4];
    const float* w_m = (const float*)d_in[5];
    const float* b_m = (const float*)d_in[6];
    const float* w_c = (const float*)d_in[7];
    float* out = (float*)d_out;
    float* ws  = (float*)d_ws;   // 13 * BSZ*H*W floats = 3.4 MB

    rect_offsets_kernel<<<(BSZ * HW) / 128, 128, 0, stream>>>(
        x, w_p, b_p, w_m, b_m, ws);

    dim3 grid(WW / 16, HH, BSZ);
    rect_main_kernel<<<grid, 128, 0, stream>>>(
        x, w_b, b_b, w_c, ws, out);
}